// GATRNN_36782099923380
// MI455X (gfx1250) — compile-verified
//
#include <hip/hip_runtime.h>
#include <hip/hip_bf16.h>
#include <stdint.h>

#define N_NODES 100000
#define N_EDGES 1600000
#define FN 32
#define FE 8
#define HID 32
#define NEG_SLOPE 0.2f

typedef __attribute__((ext_vector_type(2))) float v2f;
typedef __attribute__((ext_vector_type(8))) float v8f;

// ---------------- workspace layout (float units) ----------------
constexpr size_t OFF_H    = 0;                                   // N_NODES*HID
constexpr size_t OFF_ASRC = OFF_H    + (size_t)N_NODES * HID;    // N_NODES
constexpr size_t OFF_ADST = OFF_ASRC + N_NODES;                  // N_NODES
constexpr size_t OFF_SEGM = OFF_ADST + N_NODES;                  // N_NODES (unsigned)
constexpr size_t OFF_DEN  = OFF_SEGM + N_NODES;                  // N_NODES
constexpr size_t OFF_LOG  = OFF_DEN  + N_NODES;                  // N_EDGES + N_NODES (logits, then alpha in-place)
constexpr size_t OFF_ACC  = OFF_LOG  + (size_t)(N_EDGES + N_NODES); // N_NODES*HID
constexpr size_t OFF_MISC = OFF_ACC  + (size_t)N_NODES * HID;    // v_edge[8] + a_edge_sum[1]
// total ~8.5M floats = 34 MB

// order-preserving unsigned encoding of float (for atomicMax-based segment max)
__device__ __forceinline__ unsigned fenc(float f) {
  unsigned u = __float_as_uint(f);
  return (u & 0x80000000u) ? ~u : (u | 0x80000000u);
}
__device__ __forceinline__ float fdec(unsigned u) {
  unsigned v = (u & 0x80000000u) ? (u & 0x7FFFFFFFu) : ~u;
  return __uint_as_float(v);
}
#define ENC_NEG_INF 0x007FFFFFu   // fenc(-inf)

__device__ __forceinline__ void atomAddF(float* p, float v) { unsafeAtomicAdd(p, v); }

// ---------------- kernel 0: v_edge = W_edge @ att_edge; zero a_edge_sum ----------------
__global__ void k_setup(const float* __restrict__ W_edge, const float* __restrict__ att_edge,
                        float* __restrict__ misc) {
  int t = threadIdx.x;
  if (t < FE) {
    float s = 0.f;
    #pragma unroll
    for (int k = 0; k < HID; ++k) s += W_edge[t * HID + k] * att_edge[k];
    misc[t] = s;
  }
  if (t == FE) misc[FE] = 0.f;
}

// ---------------- kernel 1: h = x @ W via V_WMMA_F32_16X16X4_F32 ----------------
// One wave handles a 16-row tile, full 32 cols (two 16-wide N tiles), K=32 in 8 steps of 4.
__global__ void __launch_bounds__(256) k_gemm_h(const float* __restrict__ x,
                                                const float* __restrict__ W,
                                                float* __restrict__ h) {
  const int wave = blockIdx.x * (blockDim.x >> 5) + (threadIdx.x >> 5);
  const int lane = threadIdx.x & 31;
  const int ntiles = N_NODES / 16; // 6250 exactly
  if (wave >= ntiles) return;      // wave-uniform guard: EXEC stays all-1s
  const int m0 = wave * 16;
  const int r  = lane & 15;        // row (A) / col (B) within tile
  const int hi = lane >> 4;        // half-wave selects K pair

  v8f c0 = {}; v8f c1 = {};
#if __has_builtin(__builtin_amdgcn_wmma_f32_16x16x4_f32)
  const float* xrow = x + (size_t)(m0 + r) * FN;
  #pragma unroll
  for (int kc = 0; kc < 8; ++kc) {
    const int kb = kc * 4 + hi * 2;
    // A: 16x4 fp32 — lane r holds row m0+r, K=kb,kb+1 (ISA 7.12.2 layout)
    v2f a; a.x = xrow[kb]; a.y = xrow[kb + 1];
    // B: 4x16 fp32 — lane r holds column, K=kb,kb+1
    v2f b0; b0.x = W[(size_t)kb * HID + r];        b0.y = W[(size_t)(kb + 1) * HID + r];
    v2f b1; b1.x = W[(size_t)kb * HID + 16 + r];   b1.y = W[(size_t)(kb + 1) * HID + 16 + r];
    c0 = __builtin_amdgcn_wmma_f32_16x16x4_f32(false, a, false, b0, (short)0, c0, false, false);
    c1 = __builtin_amdgcn_wmma_f32_16x16x4_f32(false, a, false, b1, (short)0, c1, false, false);
  }
  // C/D layout: VGPR v holds (M = v + hi*8, N = r) for tile 0, (.., 16+r) for tile 1
  #pragma unroll
  for (int v = 0; v < 8; ++v) {
    const int row = m0 + v + hi * 8;
    h[(size_t)row * HID + r]      = c0[v];
    h[(size_t)row * HID + 16 + r] = c1[v];
  }
#else
  // scalar fallback (should not trigger on gfx1250)
  const int row = m0 + r;
  for (int n = hi * 16; n < hi * 16 + 16; ++n) {
    float s = 0.f;
    for (int k = 0; k < FN; ++k) s += x[(size_t)row * FN + k] * W[(size_t)k * HID + n];
    h[(size_t)row * HID + n] = s;
  }
#endif
}

// ---------------- kernel 2: per-node dots + init ----------------
__global__ void k_node(const float* __restrict__ h, const float* __restrict__ att_src,
                       const float* __restrict__ att_dst, float* __restrict__ a_src,
                       float* __restrict__ a_dst, unsigned* __restrict__ segmax,
                       float* __restrict__ denom, float* __restrict__ acc) {
  const int i = blockIdx.x * blockDim.x + threadIdx.x;
  if (i >= N_NODES) return;
  const float4* hp = (const float4*)(h + (size_t)i * HID);
  float s = 0.f, d = 0.f;
  #pragma unroll
  for (int q = 0; q < 8; ++q) {
    float4 hv = hp[q];
    float4 as = ((const float4*)att_src)[q];
    float4 ad = ((const float4*)att_dst)[q];
    s += hv.x * as.x + hv.y * as.y + hv.z * as.z + hv.w * as.w;
    d += hv.x * ad.x + hv.y * ad.y + hv.z * ad.z + hv.w * ad.w;
  }
  a_src[i] = s;
  a_dst[i] = d;
  segmax[i] = ENC_NEG_INF;
  denom[i] = 0.f;
  float4* ap = (float4*)(acc + (size_t)i * HID);
  float4 z = {0.f, 0.f, 0.f, 0.f};
  #pragma unroll
  for (int q = 0; q < 8; ++q) ap[q] = z;
}

// ---------------- kernel 3: edge logits + segment max + a_edge sum ----------------
__global__ void k_edge_logits(const float* __restrict__ ea, const int* __restrict__ ei,
                              const float* __restrict__ a_src, const float* __restrict__ a_dst,
                              const float* __restrict__ misc, float* __restrict__ logits,
                              unsigned* __restrict__ segmax, float* __restrict__ a_edge_sum) {
  const int e = blockIdx.x * blockDim.x + threadIdx.x;
  float ae = 0.f;
  if (e < N_EDGES) {
    const float4* ep = (const float4*)(ea + (size_t)e * FE);
    float4 e0 = ep[0], e1 = ep[1];
    ae = e0.x * misc[0] + e0.y * misc[1] + e0.z * misc[2] + e0.w * misc[3]
       + e1.x * misc[4] + e1.y * misc[5] + e1.z * misc[6] + e1.w * misc[7];
    const int s  = ei[(size_t)e * 2 + 0];
    const int dd = ei[(size_t)e * 2 + 1];
    float l = a_src[s] + a_dst[dd] + ae;
    l = (l >= 0.f) ? l : NEG_SLOPE * l;
    logits[e] = l;
    atomicMax(segmax + dd, fenc(l));
  }
  // wave-level reduction of a_edge contribution for the mean (self-loop term)
  float v = ae;
  #pragma unroll
  for (int off = 16; off > 0; off >>= 1) v += __shfl_down(v, off, 32);
  if ((threadIdx.x & 31) == 0) atomAddF(a_edge_sum, v);
}

// ---------------- kernel 4: self-loop logits (needs completed a_edge_sum) ----------------
__global__ void k_self_logits(const float* __restrict__ a_src, const float* __restrict__ a_dst,
                              const float* __restrict__ a_edge_sum, float* __restrict__ logits,
                              unsigned* __restrict__ segmax) {
  const int i = blockIdx.x * blockDim.x + threadIdx.x;
  if (i >= N_NODES) return;
  const float aem = a_edge_sum[0] * (1.0f / (float)N_EDGES);
  float l = a_src[i] + a_dst[i] + aem;
  l = (l >= 0.f) ? l : NEG_SLOPE * l;
  logits[N_EDGES + i] = l;
  atomicMax(segmax + i, fenc(l));
}

// ---------------- kernel 5: alpha = exp(l - max); denom scatter ----------------
__global__ void k_exp(const int* __restrict__ ei, float* __restrict__ logits,
                      const unsigned* __restrict__ segmax, float* __restrict__ denom) {
  const int t = blockIdx.x * blockDim.x + threadIdx.x;
  if (t >= N_EDGES + N_NODES) return;
  const int dd = (t < N_EDGES) ? ei[(size_t)t * 2 + 1] : (t - N_EDGES);
  const float m = fdec(segmax[dd]);
  const float a = __expf(logits[t] - m);
  logits[t] = a;  // alpha overwrites logit in place
  atomAddF(denom + dd, a);
}

// ---------------- kernel 6: acc[dst] += alpha * h[src] ----------------
__global__ void k_scatter(const int* __restrict__ ei, const float* __restrict__ h,
                          const float* __restrict__ alpha, float* __restrict__ acc) {
  const int t = blockIdx.x * blockDim.x + threadIdx.x;
  if (t >= N_EDGES + N_NODES) return;
  int s, dd;
  if (t < N_EDGES) { s = ei[(size_t)t * 2 + 0]; dd = ei[(size_t)t * 2 + 1]; }
  else             { s = dd = t - N_EDGES; }
  const float a = alpha[t];
  const float* hp = h + (size_t)s * HID;
  float* ap = acc + (size_t)dd * HID;
  #pragma unroll
  for (int k = 0; k < HID; ++k) atomAddF(ap + k, a * hp[k]);
}

// ---------------- kernel 7: normalize, bias, relu, final linear ----------------
__global__ void k_final(const float* __restrict__ acc, const float* __restrict__ denom,
                        const float* __restrict__ bias, const float* __restrict__ W_lin,
                        const float* __restrict__ b_lin, float* __restrict__ out) {
  const int i = blockIdx.x * blockDim.x + threadIdx.x;
  if (i >= N_NODES) return;
  const float inv = 1.0f / denom[i];
  const float* ap = acc + (size_t)i * HID;
  float y = 0.f;
  #pragma unroll
  for (int k = 0; k < HID; ++k) {
    float v = ap[k] * inv + bias[k];
    v = (v > 0.f) ? v : 0.f;
    y += v * W_lin[k];
  }
  out[i] = y + b_lin[0];
}

extern "C" void kernel_launch(void* const* d_in, const int* in_sizes, int n_in,
                              void* d_out, int out_size, void* d_ws, size_t ws_size,
                              hipStream_t stream) {
  const float* x        = (const float*)d_in[0];
  const float* ea       = (const float*)d_in[1];
  const int*   ei       = (const int*)d_in[2];
  const float* W        = (const float*)d_in[3];
  const float* att_src  = (const float*)d_in[4];
  const float* att_dst  = (const float*)d_in[5];
  const float* W_edge   = (const float*)d_in[6];
  const float* att_edge = (const float*)d_in[7];
  const float* bias     = (const float*)d_in[8];
  const float* W_lin    = (const float*)d_in[9];
  const float* b_lin    = (const float*)d_in[10];
  float* out = (float*)d_out;

  float*    ws     = (float*)d_ws;
  float*    h      = ws + OFF_H;
  float*    a_src  = ws + OFF_ASRC;
  float*    a_dst  = ws + OFF_ADST;
  unsigned* segmax = (unsigned*)(ws + OFF_SEGM);
  float*    denom  = ws + OFF_DEN;
  float*    logits = ws + OFF_LOG;
  float*    acc    = ws + OFF_ACC;
  float*    misc   = ws + OFF_MISC;           // [0..7] v_edge, [8] a_edge_sum

  const int T = 256;
  const int nb_nodes = (N_NODES + T - 1) / T;
  const int nb_edges = (N_EDGES + T - 1) / T;
  const int nb_all   = (N_EDGES + N_NODES + T - 1) / T;
  const int nb_gemm  = ((N_NODES / 16) + 7) / 8;  // 8 waves/block, 1 tile/wave

  k_setup<<<1, 32, 0, stream>>>(W_edge, att_edge, misc);
  k_gemm_h<<<nb_gemm, T, 0, stream>>>(x, W, h);
  k_node<<<nb_nodes, T, 0, stream>>>(h, att_src, att_dst, a_src, a_dst, segmax, denom, acc);
  k_edge_logits<<<nb_edges, T, 0, stream>>>(ea, ei, a_src, a_dst, misc, logits, segmax, misc + FE);
  k_self_logits<<<nb_nodes, T, 0, stream>>>(a_src, a_dst, misc + FE, logits, segmax);
  k_exp<<<nb_all, T, 0, stream>>>(ei, logits, segmax, denom);
  k_scatter<<<nb_all, T, 0, stream>>>(ei, h, logits, acc);
  k_final<<<nb_nodes, T, 0, stream>>>(acc, denom, bias, W_lin, b_lin, out);
}